// SpacialConv_15461882266185
// MI455X (gfx1250) — compile-verified
//
#include <hip/hip_runtime.h>
#include <hip/hip_bf16.h>

#define N_NODES 50000
#define N_EDGES 800000
#define IN_F    64
#define HID     4
#define OUT_F   64
#define KTOT    (IN_F + HID * IN_F)   // 64 + 256 = 320
#define EPS_N   1e-7f
#define NSLOPE  0.01f

typedef float v2f __attribute__((ext_vector_type(2)));
typedef float v8f __attribute__((ext_vector_type(8)));

// Native CDNA5 f32 atomic add (no-return form -> STOREcnt fast path in L2).
// Guarantees GLOBAL_ATOMIC_ADD_F32 instead of a possible CAS-loop expansion.
__device__ __forceinline__ void atomic_add_f32_native(float* p, float v) {
    asm volatile("global_atomic_add_f32 %0, %1, off"
                 :
                 : "v"(p), "v"(v)
                 : "memory");
}

// ---------------------------------------------------------------------------
// Zero the workspace accumulators (neigh_sum [N,256] followed by cnt [N]).
// ---------------------------------------------------------------------------
__global__ void zero_ws_kernel(float* __restrict__ p, size_t n) {
    size_t i = (size_t)blockIdx.x * blockDim.x + threadIdx.x;
    size_t stride = (size_t)gridDim.x * blockDim.x;
    for (; i < n; i += stride) p[i] = 0.0f;
}

// ---------------------------------------------------------------------------
// Edge kernel: one wave32 per edge.
//   lane handles columns j = lane + 32*t, t=0..7 (256 columns total)
//   msg[e, j] = leaky_relu(ws . W_sp[j,:] + b_sp[j]) * feat[src[e], j>>2]
//   atomically accumulated into neigh_sum[dst[e], j]; degree into cnt[dst[e]].
// Accumulator (51.2 MB) is L2-resident on MI455X (192 MB L2), so the 205M
// f32 atomic adds stay on-chip.
// ---------------------------------------------------------------------------
__global__ __launch_bounds__(256)
void edge_kernel(const float* __restrict__ feat,
                 const float* __restrict__ pos,
                 const int*   __restrict__ src,
                 const int*   __restrict__ dst,
                 const float* __restrict__ W_sp,
                 const float* __restrict__ b_sp,
                 float* __restrict__ neigh,
                 float* __restrict__ cnt) {
    const int wave = (blockIdx.x * blockDim.x + threadIdx.x) >> 5;
    const int lane = threadIdx.x & 31;
    if (wave >= N_EDGES) return;

    const int s = src[wave];
    const int d = dst[wave];

    const float rx = pos[(size_t)d * 3 + 0] - pos[(size_t)s * 3 + 0];
    const float ry = pos[(size_t)d * 3 + 1] - pos[(size_t)s * 3 + 1];
    const float rz = pos[(size_t)d * 3 + 2] - pos[(size_t)s * 3 + 2];
    const float inv = 1.0f / (sqrtf(rx * rx + ry * ry + rz * rz) + EPS_N);
    const float wx = (rx + 1.0f) * inv;
    const float wy = (ry + 1.0f) * inv;
    const float wz = (rz + 1.0f) * inv;

    const float* __restrict__ fs = feat + (size_t)s * IN_F;
    float* __restrict__ nd = neigh + (size_t)d * (HID * IN_F);

#pragma unroll
    for (int t = 0; t < 8; ++t) {
        const int j = lane + 32 * t;
        float sv = wx * W_sp[j * 3 + 0] + wy * W_sp[j * 3 + 1] +
                   wz * W_sp[j * 3 + 2] + b_sp[j];
        sv = (sv > 0.0f) ? sv : NSLOPE * sv;          // leaky_relu
        const float m = sv * fs[j >> 2];              // feat[src, j/HID]
        atomic_add_f32_native(&nd[j], m);
    }
    if (lane == 0) atomic_add_f32_native(&cnt[d], 1.0f);
}

// ---------------------------------------------------------------------------
// Node GEMM kernel (WMMA f32 16x16x4):
//   out[n,o] = sum_k A[n,k] * B[k,o] + bias[o] + b_neigh[o]
//   A[n,k] = feat[n,k]                     for k <  64
//          = neigh_sum[n,k-64]/max(cnt,1)  for k >= 64
//   B[k,o] = W_self[o,k]                   for k <  64
//          = W_neigh[o,k-64]               for k >= 64
// Block = 128 threads = 4 waves; wave w handles output cols [16w,16w+16);
// block handles 16 rows. 50000/16 = 3125 blocks; 320/4 = 80 WMMAs per wave.
//
// VGPR layouts per CDNA5 ISA (wave32):
//   A 16x4 f32 : lanes 0-15 -> K={k0,k0+1}, lanes 16-31 -> K={k0+2,k0+3}; M = lane&15
//   B 4x16 f32 : same K split; N = lane&15
//   C 16x16 f32: VGPR r -> M = r + 8*(lane>=16); N = lane&15
// ---------------------------------------------------------------------------
__global__ __launch_bounds__(128)
void node_gemm_kernel(const float* __restrict__ feat,
                      const float* __restrict__ neigh,
                      const float* __restrict__ cnt,
                      const float* __restrict__ W_self,
                      const float* __restrict__ W_neigh,
                      const float* __restrict__ b_neigh,
                      const float* __restrict__ bias,
                      float* __restrict__ out) {
    const int lane   = threadIdx.x & 31;
    const int wv     = threadIdx.x >> 5;     // 0..3: output column tile
    const int half   = lane >> 4;            // K-subgroup (0: K+0/1, 1: K+2/3)
    const int l15    = lane & 15;
    const int m_base = blockIdx.x * 16;
    const int row    = m_base + l15;         // A row handled by this lane
    const int col    = wv * 16 + l15;        // B/D column handled by this lane

    const float inv = 1.0f / fmaxf(cnt[row], 1.0f);
    const float* __restrict__ fr = feat  + (size_t)row * IN_F;
    const float* __restrict__ nr = neigh + (size_t)row * (HID * IN_F);
    const float* __restrict__ ws = W_self  + (size_t)col * IN_F;
    const float* __restrict__ wn = W_neigh + (size_t)col * (HID * IN_F);

    v8f acc = {};

    // K = 0..63 : feat x W_self^T
#pragma unroll
    for (int k0 = 0; k0 < IN_F; k0 += 4) {
        const int k = k0 + 2 * half;
        v2f a; a[0] = fr[k];  a[1] = fr[k + 1];
        v2f b; b[0] = ws[k];  b[1] = ws[k + 1];
        acc = __builtin_amdgcn_wmma_f32_16x16x4_f32(
                  false, a, false, b, (short)0, acc, false, false);
    }
    // K = 64..319 : (neigh_sum/cnt) x W_neigh^T
#pragma unroll 4
    for (int k0 = 0; k0 < HID * IN_F; k0 += 4) {
        const int k = k0 + 2 * half;
        v2f a; a[0] = nr[k] * inv;  a[1] = nr[k + 1] * inv;
        v2f b; b[0] = wn[k];        b[1] = wn[k + 1];
        acc = __builtin_amdgcn_wmma_f32_16x16x4_f32(
                  false, a, false, b, (short)0, acc, false, false);
    }

    const float bb = bias[col] + b_neigh[col];
#pragma unroll
    for (int r = 0; r < 8; ++r) {
        const int m = m_base + half * 8 + r;
        out[(size_t)m * OUT_F + col] = acc[r] + bb;
    }
}

// ---------------------------------------------------------------------------
extern "C" void kernel_launch(void* const* d_in, const int* in_sizes, int n_in,
                              void* d_out, int out_size, void* d_ws, size_t ws_size,
                              hipStream_t stream) {
    const float* feat    = (const float*)d_in[0];
    const float* pos     = (const float*)d_in[1];
    const int*   src     = (const int*)  d_in[2];
    const int*   dst     = (const int*)  d_in[3];
    const float* W_self  = (const float*)d_in[4];
    const float* W_sp    = (const float*)d_in[5];
    const float* b_sp    = (const float*)d_in[6];
    const float* W_neigh = (const float*)d_in[7];
    const float* b_neigh = (const float*)d_in[8];
    const float* bias    = (const float*)d_in[9];
    float* out = (float*)d_out;

    float* neigh = (float*)d_ws;                         // [N, 256]
    float* cnt   = neigh + (size_t)N_NODES * HID * IN_F; // [N]

    const size_t zcount = (size_t)N_NODES * HID * IN_F + N_NODES;
    zero_ws_kernel<<<4096, 256, 0, stream>>>(neigh, zcount);

    // one wave32 per edge, 8 edges per 256-thread block
    edge_kernel<<<N_EDGES / 8, 256, 0, stream>>>(
        feat, pos, src, dst, W_sp, b_sp, neigh, cnt);

    // fused node GEMM: 3125 blocks x (4 waves of 16x16 tiles)
    node_gemm_kernel<<<N_NODES / 16, 128, 0, stream>>>(
        feat, neigh, cnt, W_self, W_neigh, b_neigh, bias, out);
}